// Attention_2199023255769
// MI455X (gfx1250) — compile-verified
//
#include <hip/hip_runtime.h>
#include <hip/hip_bf16.h>
#include <math.h>

// Problem constants (match the reference)
#define B_   4
#define SEQ  2048     // Q == K == 2048
#define CDIM 128
#define HEADS 8
#define DDIM 16

#define QTILES 4      // 16-row sub-tiles per wave in the attention kernel (64 rows)

typedef float v2f __attribute__((ext_vector_type(2)));
typedef float v8f __attribute__((ext_vector_type(8)));

// ---------------------------------------------------------------------------
// Kernel 1: fused projection.  One wave (32 threads) computes one 16(seq) x
// 16(d) output tile for one head, for one of {q,k,v} (blockIdx.z).
//   y[s, d] = sum_c x[s,c] * w[h*16+d, c]      (q additionally * 1/sqrt(D))
// q,k stored transposed [B,H,D,S] (for coalesced A/B fragment loads later),
// v stored [B,H,S,D].
// ---------------------------------------------------------------------------
__global__ __launch_bounds__(32) void attn_proj_kernel(
    const float* __restrict__ q_x, const float* __restrict__ kv_x,
    const float* __restrict__ w_q, const float* __restrict__ w_k,
    const float* __restrict__ w_v,
    float* __restrict__ qT, float* __restrict__ kT, float* __restrict__ vO)
{
    const int lane = threadIdx.x;
    const int half = lane >> 4;      // 0 or 1 (lane group)
    const int l16  = lane & 15;
    const int s0   = blockIdx.x * 16;      // seq tile start
    const int bh   = blockIdx.y;           // b*HEADS + h
    const int h    = bh % HEADS;
    const int b    = bh / HEADS;
    const int mode = blockIdx.z;           // 0=q, 1=k, 2=v

    const float* x = (mode == 0) ? q_x : kv_x;
    const float* w = (mode == 0) ? w_q : (mode == 1 ? w_k : w_v);

    __shared__ float xs[16 * 132];   // padded stride: conflict-free frag reads
    __shared__ float ws[16 * 132];
    __shared__ float ts[16 * 20];    // transpose bounce for q/k stores

    // Cooperative, coalesced load of the 16x128 activation and weight tiles.
    {
        const float4* xg = (const float4*)(x + (size_t)b * SEQ * CDIM
                                             + (size_t)s0 * CDIM);
        const float4* wg = (const float4*)(w + (size_t)h * 16 * CDIM);
        #pragma unroll
        for (int r = 0; r < 16; ++r) {
            float4 xv = xg[r * 32 + lane];
            float4 wv = wg[r * 32 + lane];
            *(float4*)(&xs[r * 132 + lane * 4]) = xv;
            *(float4*)(&ws[r * 132 + lane * 4]) = wv;
        }
    }
    __syncthreads();

    // 16x16x128 GEMM as 32 chained f32 WMMAs (K=4 each).
    // A frag (16x4 f32): lane holds row M=l16, K = 2*half + j.
    // B frag (4x16):     lane holds col N=l16, K = 2*half + j.
    v8f acc = {};
    #pragma unroll
    for (int c0 = 0; c0 < CDIM; c0 += 4) {
        const int cc = c0 + 2 * half;
        v2f a, bb;
        a.x  = xs[l16 * 132 + cc + 0];
        a.y  = xs[l16 * 132 + cc + 1];
        bb.x = ws[l16 * 132 + cc + 0];
        bb.y = ws[l16 * 132 + cc + 1];
        acc = __builtin_amdgcn_wmma_f32_16x16x4_f32(
                  false, a, false, bb, (short)0, acc, false, false);
    }

    if (mode == 2) {
        // v: [B,H,S,D] row-major; C/D layout row m = j + 8*half, col n = l16.
        float* vb = vO + (size_t)bh * SEQ * DDIM;
        #pragma unroll
        for (int j = 0; j < 8; ++j)
            vb[(size_t)(s0 + j + 8 * half) * DDIM + l16] = acc[j];
    } else {
        // q/k: transpose through LDS, store [B,H,D,S] coalesced.
        const float sc = (mode == 0) ? 0.25f : 1.0f;   // 1/sqrt(D) for q
        __syncthreads();
        #pragma unroll
        for (int j = 0; j < 8; ++j)
            ts[(j + 8 * half) * 20 + l16] = acc[j] * sc;
        __syncthreads();
        float* base = ((mode == 0) ? qT : kT) + (size_t)bh * DDIM * SEQ;
        #pragma unroll
        for (int j = 0; j < 8; ++j) {
            const int d = j + 8 * half;
            base[(size_t)d * SEQ + s0 + l16] = ts[l16 * 20 + d];
        }
    }
}

// ---------------------------------------------------------------------------
// Kernel 2: flash attention.  One wave owns a 64-row Q tile (QTILES stacked
// 16x16 accumulators) of one (b,h).  Per 16-wide K step:
//   - K-tile B fragments loaded ONCE, reused by 4 S-WMMAs (one per sub-tile)
//   - online softmax per sub-tile, fully in registers (intra-half shuffles)
//   - P sub-tiles bounced through private LDS buffers (C/D -> A layout)
//   - V-tile B fragments loaded ONCE, reused by 4 PV-WMMA chains
// => 32 WMMAs per K step against 16 fragment loads + 2 barriers.
// Output: [B, Q, H, D].
// ---------------------------------------------------------------------------
__global__ __launch_bounds__(32) void attn_flash_kernel(
    const float* __restrict__ qT, const float* __restrict__ kT,
    const float* __restrict__ vO, float* __restrict__ out)
{
    const int lane = threadIdx.x;
    const int half = lane >> 4;
    const int l16  = lane & 15;
    const int q0   = blockIdx.x * (16 * QTILES);
    const int bh   = blockIdx.y;
    const int h    = bh % HEADS;
    const int b    = bh / HEADS;

    const float* qTh = qT + (size_t)bh * DDIM * SEQ;
    const float* kTh = kT + (size_t)bh * DDIM * SEQ;
    const float* vh  = vO + (size_t)bh * SEQ * DDIM;

    __shared__ float sP[QTILES][16 * 20];   // per-sub-tile P bounce (padded)

    // Q A-fragments for all sub-tiles and all 4 d-chunks: resident all loop.
    v2f qa[QTILES][4];
    #pragma unroll
    for (int t = 0; t < QTILES; ++t)
        #pragma unroll
        for (int kk = 0; kk < 4; ++kk) {
            const int d0 = kk * 4 + 2 * half;
            qa[t][kk].x = qTh[(size_t)(d0 + 0) * SEQ + q0 + t * 16 + l16];
            qa[t][kk].y = qTh[(size_t)(d0 + 1) * SEQ + q0 + t * 16 + l16];
        }

    v8f acc_o[QTILES];
    float rm[QTILES][8], rl[QTILES][8];
    #pragma unroll
    for (int t = 0; t < QTILES; ++t) {
        acc_o[t] = (v8f){};
        #pragma unroll
        for (int j = 0; j < 8; ++j) { rm[t][j] = -1e30f; rl[t][j] = 0.0f; }
    }

    for (int k0 = 0; k0 < SEQ; k0 += 16) {
        // ---- K-tile B fragments: load once, reuse for all sub-tiles ----
        v2f kb[4];
        #pragma unroll
        for (int kk = 0; kk < 4; ++kk) {
            const int d0 = kk * 4 + 2 * half;
            kb[kk].x = kTh[(size_t)(d0 + 0) * SEQ + k0 + l16];
            kb[kk].y = kTh[(size_t)(d0 + 1) * SEQ + k0 + l16];
        }
        // ---- V-tile B fragments: load once, reuse for all sub-tiles ----
        v2f vb[4];
        #pragma unroll
        for (int kk = 0; kk < 4; ++kk) {
            const int kr = kk * 4 + 2 * half;
            vb[kk].x = vh[(size_t)(k0 + kr + 0) * DDIM + l16];
            vb[kk].y = vh[(size_t)(k0 + kr + 1) * DDIM + l16];
        }

        __syncthreads();   // sP reads of previous iteration are done

        #pragma unroll
        for (int t = 0; t < QTILES; ++t) {
            // S_t = Q_t (16x16) x K^T tile (16x16), K-dim = d.
            v8f s = {};
            #pragma unroll
            for (int kk = 0; kk < 4; ++kk)
                s = __builtin_amdgcn_wmma_f32_16x16x4_f32(
                        false, qa[t][kk], false, kb[kk], (short)0, s,
                        false, false);

            // Online softmax.  Row m = j + 8*half lives on 16 lanes of one
            // half, so masks 1/2/4/8 reduce within the row.
            #pragma unroll
            for (int j = 0; j < 8; ++j) {
                float v  = s[j];
                float mx = v;
                mx = fmaxf(mx, __shfl_xor(mx, 1, 32));
                mx = fmaxf(mx, __shfl_xor(mx, 2, 32));
                mx = fmaxf(mx, __shfl_xor(mx, 4, 32));
                mx = fmaxf(mx, __shfl_xor(mx, 8, 32));
                const float mnew  = fmaxf(rm[t][j], mx);
                const float scale = __expf(rm[t][j] - mnew);
                const float pv    = __expf(v - mnew);
                float rs = pv;
                rs += __shfl_xor(rs, 1, 32);
                rs += __shfl_xor(rs, 2, 32);
                rs += __shfl_xor(rs, 4, 32);
                rs += __shfl_xor(rs, 8, 32);
                rl[t][j]    = rl[t][j] * scale + rs;
                rm[t][j]    = mnew;
                acc_o[t][j] = acc_o[t][j] * scale;
                // C/D layout -> LDS [row m][col n] for the A-layout reload.
                sP[t][(j + 8 * half) * 20 + l16] = pv;
            }
        }

        __syncthreads();   // all P sub-tiles visible

        // O_t += P_t (16x16, K = k-index) x V (16x16): vb reused 4x.
        #pragma unroll
        for (int t = 0; t < QTILES; ++t) {
            #pragma unroll
            for (int kk = 0; kk < 4; ++kk) {
                const int kr = kk * 4 + 2 * half;
                v2f pa;
                pa.x = sP[t][l16 * 20 + kr + 0];
                pa.y = sP[t][l16 * 20 + kr + 1];
                acc_o[t] = __builtin_amdgcn_wmma_f32_16x16x4_f32(
                               false, pa, false, vb[kk], (short)0, acc_o[t],
                               false, false);
            }
        }
    }

    // Epilogue: normalize and store [B, Q, H, D] (coalesced per half-wave).
    #pragma unroll
    for (int t = 0; t < QTILES; ++t)
        #pragma unroll
        for (int j = 0; j < 8; ++j) {
            const int m = q0 + t * 16 + j + 8 * half;
            out[(((size_t)b * SEQ + m) * HEADS + h) * DDIM + l16] =
                acc_o[t][j] / rl[t][j];
        }
}

// ---------------------------------------------------------------------------
extern "C" void kernel_launch(void* const* d_in, const int* in_sizes, int n_in,
                              void* d_out, int out_size, void* d_ws, size_t ws_size,
                              hipStream_t stream)
{
    const float* q_x  = (const float*)d_in[0];
    const float* kv_x = (const float*)d_in[1];
    const float* w_q  = (const float*)d_in[2];
    const float* w_k  = (const float*)d_in[3];
    const float* w_v  = (const float*)d_in[4];
    float* out = (float*)d_out;

    // Workspace: qT [B,H,D,S] + kT [B,H,D,S] + v [B,H,S,D]  (3 x 4 MB)
    const size_t perTensor = (size_t)B_ * HEADS * DDIM * SEQ;
    float* qT = (float*)d_ws;
    float* kT = qT + perTensor;
    float* vO = kT + perTensor;

    dim3 gProj(SEQ / 16, B_ * HEADS, 3);
    attn_proj_kernel<<<gProj, 32, 0, stream>>>(q_x, kv_x, w_q, w_k, w_v,
                                               qT, kT, vO);

    dim3 gAttn(SEQ / (16 * QTILES), B_ * HEADS);
    attn_flash_kernel<<<gAttn, 32, 0, stream>>>(qT, kT, vO, out);
}